// CustomFullyConnectedLayerSoftmax_65618510348676
// MI455X (gfx1250) — compile-verified
//
#include <hip/hip_runtime.h>
#include <math.h>

// Problem constants (from reference): IN_F = OUT_F = TOTAL = DIAG = 2048, BATCH = 32
#define NF     2048
#define BATCH  32
#define NT     64        // output-column tile per workgroup
#define KB     128       // k-chunk staged in LDS (must be power of 2; shift = 7)
#define KSOFT  1844.0f   // ceil(int(0.9*2048*2048)/2048)
#define LDX    (KB + 1)  // padded LDS stride for x tile
#define LDB    (NT + 1)  // padded LDS stride for B tile
#define KSPLIT 8

typedef float v2f __attribute__((ext_vector_type(2)));
typedef float v8f __attribute__((ext_vector_type(8)));

// ---------------------------------------------------------------------------
// s[t] = clip(KSOFT * softmax(alpha)[t], 0, 1), single workgroup over 2048.
// ---------------------------------------------------------------------------
__global__ __launch_bounds__(256)
void softmax_scale_kernel(const float* __restrict__ alpha, float* __restrict__ s) {
    __shared__ float red[256];
    const int tid = threadIdx.x;

    float mx = -3.4e38f;
    for (int i = tid; i < NF; i += 256) mx = fmaxf(mx, alpha[i]);
    red[tid] = mx;
    __syncthreads();
    for (int off = 128; off > 0; off >>= 1) {
        if (tid < off) red[tid] = fmaxf(red[tid], red[tid + off]);
        __syncthreads();
    }
    const float m = red[0];
    __syncthreads();

    float sm = 0.0f;
    for (int i = tid; i < NF; i += 256) sm += expf(alpha[i] - m);
    red[tid] = sm;
    __syncthreads();
    for (int off = 128; off > 0; off >>= 1) {
        if (tid < off) red[tid] += red[tid + off];
        __syncthreads();
    }
    const float inv = 1.0f / red[0];

    for (int i = tid; i < NF; i += 256) {
        float v = KSOFT * expf(alpha[i] - m) * inv;
        s[i] = fminf(1.0f, fmaxf(0.0f, v));
    }
}

// ---------------------------------------------------------------------------
// Diagonal-gather GEMM: out[b,n] = sum_k x[b,k] * V[(n-k)&2047, k] * s[(n-k)&2047]
// Grid: (NF/NT, nsplit). Each block: 256 threads = 8 waves, each wave owns one
// 16x16 f32 WMMA accumulator tile of the 32x64 block tile.
// ---------------------------------------------------------------------------
__global__ __launch_bounds__(256)
void diag_gemm_kernel(const float* __restrict__ x, const float* __restrict__ V,
                      const float* __restrict__ s, float* __restrict__ outp,
                      int chunksPerSplit) {
    __shared__ float x_s[BATCH * LDX];        // 32 x 128 x-tile
    __shared__ float b_s[KB * LDB];           // 128 x 64 B tile (B[k][n] = W^T)
    __shared__ float s_s[NT + KB];            // staged scale window (191 used)

    const int tid  = threadIdx.x;
    const int lane = tid & 31;
    const int wave = tid >> 5;
    const int mi   = wave >> 2;               // 0..1 : batch-row 16-tile
    const int ni   = wave & 3;                // 0..3 : out-col  16-tile
    const int hi   = lane >> 4;               // lane half (selects K/M half)
    const int l15  = lane & 15;

    const int n0 = blockIdx.x * NT;
    const int ks = blockIdx.y;
    float* dst = outp + (size_t)ks * (BATCH * NF);

    v8f acc = {};

    for (int c = 0; c < chunksPerSplit; ++c) {
        const int k0 = (ks * chunksPerSplit + c) * KB;
        __syncthreads();   // protect LDS reuse across chunks

        // Stage x tile: x_s[r][j] = x[r, k0+j]  (coalesced rows)
        for (int i = tid; i < BATCH * KB; i += 256) {
            const int r   = i >> 7;           // i / KB
            const int col = i & (KB - 1);
            x_s[r * LDX + col] = x[r * NF + k0 + col];
        }

        // Stage scale window: diagonals t = n0-k0 + (n-j), n-j in [-(KB-1), NT-1]
        const int t0 = (n0 - k0 - (KB - 1)) & (NF - 1);
        for (int i = tid; i < NT + KB - 1; i += 256)
            s_s[i] = s[(t0 + i) & (NF - 1)];
        __syncthreads();

        // Stage B tile via diagonal walk: along (j+1, n+1) the V address is +1,
        // so runs of up to 64 consecutive threads read 256 contiguous bytes of one V row.
        for (int it = 0; it < (KB * NT) / 256; ++it) {
            const int idx = it * 256 + tid;
            const int j   = idx & (KB - 1);           // local k
            const int g   = idx >> 7;                 // diagonal id 0..NT-1
            const int n   = (g + j) & (NT - 1);       // local n
            const int tl  = n - j + (KB - 1);         // 0 .. NT+KB-2
            const int t   = (t0 + tl) & (NF - 1);     // V row (diagonal index)
            b_s[j * LDB + n] = V[(size_t)t * NF + (k0 + j)] * s_s[tl];
        }
        __syncthreads();

        // WMMA loop: D += A(16x4) * B(4x16), exact f32 math.
        // A frag: VGPR0 = K{+0 lo / +2 hi}, VGPR1 = K{+1 lo / +3 hi}
        // B frag: VGPR0 = row K(+2*hi),  VGPR1 = row K(+2*hi+1), col = lane&15
        for (int kk = 0; kk < KB; kk += 4) {
            const int ka = kk + 2 * hi;
            v2f a, b;
            a.x = x_s[(mi * 16 + l15) * LDX + ka];
            a.y = x_s[(mi * 16 + l15) * LDX + ka + 1];
            b.x = b_s[ka * LDB + ni * 16 + l15];
            b.y = b_s[(ka + 1) * LDB + ni * 16 + l15];
            acc = __builtin_amdgcn_wmma_f32_16x16x4_f32(
                /*neg_a=*/false, a, /*neg_b=*/false, b,
                /*c_mod=*/(short)0, acc, /*reuse_a=*/false, /*reuse_b=*/false);
        }
    }

    // C/D layout: VGPR r -> M = r + 8*(lane>=16), N = lane&15
    const int row = mi * 16 + hi * 8;
    const int col = n0 + ni * 16 + l15;
#pragma unroll
    for (int r = 0; r < 8; ++r)
        dst[(size_t)(row + r) * NF + col] = acc[r];
}

// ---------------------------------------------------------------------------
// Deterministic reduction of K-split partials (no float atomics).
// ---------------------------------------------------------------------------
__global__ __launch_bounds__(256)
void reduce_kernel(const float* __restrict__ part, float* __restrict__ out, int nsplit) {
    const int i = blockIdx.x * 256 + threadIdx.x;   // < BATCH*NF
    float v = 0.0f;
    for (int p = 0; p < nsplit; ++p)
        v += part[(size_t)p * (BATCH * NF) + i];
    out[i] = v;
}

extern "C" void kernel_launch(void* const* d_in, const int* in_sizes, int n_in,
                              void* d_out, int out_size, void* d_ws, size_t ws_size,
                              hipStream_t stream) {
    (void)in_sizes; (void)n_in; (void)out_size;
    const float* x     = (const float*)d_in[0];   // [32, 2048] f32
    const float* V     = (const float*)d_in[1];   // [2048, 2048] f32
    const float* alpha = (const float*)d_in[2];   // [2048] f32
    float* out = (float*)d_out;                   // [32, 2048] f32

    float* s = (float*)d_ws;                      // 2048 f32 = 8 KB
    const size_t sBytes    = (size_t)NF * sizeof(float);
    const size_t partBytes = (size_t)KSPLIT * BATCH * NF * sizeof(float);

    softmax_scale_kernel<<<1, 256, 0, stream>>>(alpha, s);

    if (ws_size >= sBytes + partBytes) {
        // K split 8 ways -> 256 workgroups for bandwidth parallelism.
        float* part = (float*)((char*)d_ws + sBytes);
        diag_gemm_kernel<<<dim3(NF / NT, KSPLIT), 256, 0, stream>>>(
            x, V, s, part, (NF / KB) / KSPLIT);
        reduce_kernel<<<dim3((BATCH * NF) / 256), 256, 0, stream>>>(part, out, KSPLIT);
    } else {
        // Fallback: each block owns full K for its n-tile, writes out directly.
        diag_gemm_kernel<<<dim3(NF / NT, 1), 256, 0, stream>>>(
            x, V, s, out, NF / KB);
    }
}